// Tsi_model_56994216018169
// MI455X (gfx1250) — compile-verified
//
#include <hip/hip_runtime.h>

typedef __attribute__((ext_vector_type(2))) float v2f;
typedef __attribute__((ext_vector_type(8))) float v8f;

#define N_NODES 50000
#define IN_FEAT 128
#define MTILES 5   // 50000/16 = 3125 = 5 * 625 row tiles -> exact, no bounds checks

// ---------------------------------------------------------------------------
// Degree: deg[d] += 1 per edge (float atomics; exact for counts < 2^24)
// ---------------------------------------------------------------------------
__global__ void degree_kernel(const int* __restrict__ dst, float* __restrict__ deg, int nedges) {
    int e = blockIdx.x * blockDim.x + threadIdx.x;
    if (e < nedges) atomicAdd(&deg[dst[e]], 1.0f);
}

// dinv = rsqrt(deg + 1)  (self-loop included), in place
__global__ void dinv_kernel(float* __restrict__ deg, int n) {
    int i = blockIdx.x * blockDim.x + threadIdx.x;
    if (i < n) deg[i] = rsqrtf(deg[i] + 1.0f);
}

// ---------------------------------------------------------------------------
// C[M x NOUT] = A[M x 128] * B[128 x NOUT] via V_WMMA_F32_16X16X4_F32.
// One wave owns one 16-wide column stripe; its full B panel (128x16 f32) is
// held register-resident (32 x v2f = 64 VGPRs under wave32), then reused
// across MTILES row tiles -> steady state is 1 v2f A-load + 1 WMMA per k-step.
// Wave32 layouts (ISA 7.12.2):
//   A 16x4 f32:  lane = m | m+16;  VGPR0 = K{0|2}, VGPR1 = K{1|3}
//   B  4x16 f32: lane = n | n+16;  VGPR0 = K{0|2}, VGPR1 = K{1|3}
//   C/D 16x16:   VGPR r: lanes0-15 -> M=r, lanes16-31 -> M=r+8; N = lane&15
// ---------------------------------------------------------------------------
template <int NOUT>
__global__ void __launch_bounds__((NOUT / 16) * 32)
gemm_wmma_f32(const float* __restrict__ A, const float* __restrict__ B, float* __restrict__ C) {
    const int K = IN_FEAT;
    const int wave = threadIdx.x >> 5;
    const int lane = threadIdx.x & 31;
    const int mn   = lane & 15;      // row (A) / col (B) within tile
    const int kh   = lane >> 4;      // K half-select (0 -> K=0,1 ; 1 -> K=2,3)

    const int col0 = wave << 4;

    // ---- Load register-resident B panel: 32 k-steps x v2f ----
    const float* __restrict__ bbase = B + col0 + mn;
    v2f bfrag[K / 4];
#pragma unroll
    for (int kk = 0; kk < K / 4; ++kk) {
        const int k = kk * 4 + 2 * kh;
        v2f b;
        b.x = bbase[(size_t)k * NOUT];
        b.y = bbase[(size_t)(k + 1) * NOUT];
        bfrag[kk] = b;
    }

    // ---- Sweep MTILES row tiles reusing bfrag ----
    for (int t = 0; t < MTILES; ++t) {
        const int row0 = (blockIdx.x * MTILES + t) << 4;
        const float* __restrict__ arow = A + (size_t)(row0 + mn) * K + 2 * kh;

        v8f acc = {};
#pragma unroll
        for (int kk = 0; kk < K / 4; ++kk) {
            v2f a = *(const v2f*)(arow + kk * 4);   // K = 4kk+2kh, +1 (8B aligned)
            acc = __builtin_amdgcn_wmma_f32_16x16x4_f32(
                /*neg_a=*/false, a, /*neg_b=*/false, bfrag[kk],
                /*c_mod=*/(short)0, acc, /*reuse_a=*/false, /*reuse_b=*/false);
        }

#pragma unroll
        for (int r = 0; r < 8; ++r) {
            C[(size_t)(row0 + r + 8 * kh) * NOUT + col0 + mn] = acc[r];
        }
    }
}

// ---------------------------------------------------------------------------
// Edge aggregation: agg[dst] += h[src] * dinv[src]*dinv[dst]
// One lane per (edge, feature): coalesced 512B gather + f32 atomic scatter
// (both L2-resident: feature matrix is 25.6 MB << 192 MB L2).
// ---------------------------------------------------------------------------
template <int F>
__global__ void gcn_aggregate(const float* __restrict__ h, const int* __restrict__ src,
                              const int* __restrict__ dst, const float* __restrict__ dinv,
                              float* __restrict__ agg, int nedges) {
    const int EPB = 256 / F;
    int e = blockIdx.x * EPB + (int)(threadIdx.x / F);
    if (e >= nedges) return;
    int f = threadIdx.x & (F - 1);
    int s = src[e];
    int d = dst[e];
    float w = dinv[s] * dinv[d];
    atomicAdd(&agg[(size_t)d * F + f], h[(size_t)s * F + f] * w);
}

// ---------------------------------------------------------------------------
// Fused: agg += h_pre * dinv^2 (self loop) + bias, optional SELU. In place.
// ---------------------------------------------------------------------------
template <int F, bool SELU>
__global__ void finalize_kernel(float* __restrict__ agg, const float* __restrict__ hpre,
                                const float* __restrict__ dinv, const float* __restrict__ bias,
                                int total) {
    int i = blockIdx.x * blockDim.x + threadIdx.x;
    if (i >= total) return;
    int node = i / F;                 // F is power of two -> shift
    int f    = i & (F - 1);
    float di = dinv[node];
    float v  = agg[i] + hpre[i] * di * di + bias[f];
    if (SELU) {
        const float alpha = 1.6732632423543772848f;
        const float scale = 1.0507009873554804934f;
        v = scale * (v > 0.0f ? v : alpha * (expf(v) - 1.0f));
    }
    agg[i] = v;
}

// ---------------------------------------------------------------------------
// Row softmax over 64 features: one wave per node, 2 values per lane,
// wave32 shuffle reductions.
// ---------------------------------------------------------------------------
__global__ void softmax64_kernel(const float* __restrict__ in, float* __restrict__ out, int n) {
    int node = blockIdx.x * (blockDim.x >> 5) + (threadIdx.x >> 5);
    int lane = threadIdx.x & 31;
    if (node >= n) return;
    size_t base = (size_t)node * 64;
    float v0 = in[base + lane];
    float v1 = in[base + lane + 32];
    float m = fmaxf(v0, v1);
#pragma unroll
    for (int off = 16; off > 0; off >>= 1) m = fmaxf(m, __shfl_xor(m, off, 32));
    float e0 = expf(v0 - m);
    float e1 = expf(v1 - m);
    float s = e0 + e1;
#pragma unroll
    for (int off = 16; off > 0; off >>= 1) s += __shfl_xor(s, off, 32);
    float inv = 1.0f / s;
    out[base + lane]      = e0 * inv;
    out[base + lane + 32] = e1 * inv;
}

// ---------------------------------------------------------------------------
// Host launcher
// ---------------------------------------------------------------------------
extern "C" void kernel_launch(void* const* d_in, const int* in_sizes, int n_in,
                              void* d_out, int out_size, void* d_ws, size_t ws_size,
                              hipStream_t stream) {
    const float* x  = (const float*)d_in[0];
    const int*   ei = (const int*)d_in[1];
    const float* W1 = (const float*)d_in[2];
    const float* b1 = (const float*)d_in[3];
    const float* W2 = (const float*)d_in[4];
    const float* b2 = (const float*)d_in[5];

    const int E = in_sizes[1] / 2;       // edge_index is [2, E] flat
    const int* src = ei;
    const int* dst = ei + E;

    // Workspace carve-out (256B aligned)
    char*  ws  = (char*)d_ws;
    size_t off = 0;
    auto carve = [&](size_t bytes) -> void* {
        void* p = ws + off;
        off += (bytes + 255) & ~(size_t)255;
        return p;
    };
    float* dinv = (float*)carve((size_t)N_NODES * 4);
    float* h1p  = (float*)carve((size_t)N_NODES * 128 * 4);
    float* agg1 = (float*)carve((size_t)N_NODES * 128 * 4);
    float* h2p  = (float*)carve((size_t)N_NODES * 64 * 4);
    float* agg2 = (float*)carve((size_t)N_NODES * 64 * 4);

    // Zero accumulation buffers every call (graph-capture safe)
    hipMemsetAsync(dinv, 0, (size_t)N_NODES * 4, stream);
    hipMemsetAsync(agg1, 0, (size_t)N_NODES * 128 * 4, stream);
    hipMemsetAsync(agg2, 0, (size_t)N_NODES * 64 * 4, stream);

    // Degree + symmetric normalization
    degree_kernel<<<(E + 255) / 256, 256, 0, stream>>>(dst, dinv, E);
    dinv_kernel<<<(N_NODES + 255) / 256, 256, 0, stream>>>(dinv, N_NODES);

    // ---- Layer 1: h1 = selu(GCNConv(x, W1, b1)) ----
    gemm_wmma_f32<128><<<N_NODES / (16 * MTILES), 256, 0, stream>>>(x, W1, h1p);
    gcn_aggregate<128><<<(E + 1) / 2, 256, 0, stream>>>(h1p, src, dst, dinv, agg1, E);
    finalize_kernel<128, true><<<((N_NODES * 128) + 255) / 256, 256, 0, stream>>>(
        agg1, h1p, dinv, b1, N_NODES * 128);

    // ---- Layer 2: h2 = GCNConv(h1, W2, b2); softmax ----
    gemm_wmma_f32<64><<<N_NODES / (16 * MTILES), 128, 0, stream>>>(agg1, W2, h2p);
    gcn_aggregate<64><<<(E + 3) / 4, 256, 0, stream>>>(h2p, src, dst, dinv, agg2, E);
    finalize_kernel<64, false><<<((N_NODES * 64) + 255) / 256, 256, 0, stream>>>(
        agg2, h2p, dinv, b2, N_NODES * 64);

    softmax64_kernel<<<(N_NODES + 7) / 8, 256, 0, stream>>>(agg2, (float*)d_out, N_NODES);
}